// v8loss_67645734912001
// MI455X (gfx1250) — compile-verified
//
#include <hip/hip_runtime.h>
#include <math.h>

#define BS 16
#define NA 8400
#define NB 64
#define NC 80
#define KTOP 10
#define EPSF 1e-9f

typedef float v2f __attribute__((ext_vector_type(2)));
typedef float v8f __attribute__((ext_vector_type(8)));

// ---------------------------------------------------------------------------
// Clipped CIoU (matches reference ciou_xyxy followed by clip(.,0)).
// area/atan of each box are precomputed by the caller.
// ---------------------------------------------------------------------------
__device__ __forceinline__ float ciou_clipped(
    float x1, float y1, float x2, float y2, float area1, float at1,
    float gx1, float gy1, float gx2, float gy2, float area2, float at2)
{
    float iw    = fminf(x2, gx2) - fmaxf(x1, gx1);
    float ih    = fminf(y2, gy2) - fmaxf(y1, gy1);
    float inter = fmaxf(iw, 0.f) * fmaxf(ih, 0.f);
    float uni   = area1 + area2 - inter + 1e-7f;
    float iou   = inter / uni;
    float cw    = fmaxf(x2, gx2) - fminf(x1, gx1);
    float ch    = fmaxf(y2, gy2) - fminf(y1, gy1);
    float c2    = cw * cw + ch * ch + 1e-7f;
    float dx    = gx1 + gx2 - x1 - x2;
    float dy    = gy1 + gy2 - y1 - y2;
    float rho2  = (dx * dx + dy * dy) * 0.25f;
    float dv    = at2 - at1;
    float vv    = (4.0f / (float)(M_PI * M_PI)) * dv * dv;
    float alpha = vv / (vv - iou + 1.0f + 1e-7f);
    float ciou  = iou - (rho2 / c2 + vv * alpha);
    return fmaxf(ciou, 0.f);
}

// ---------------------------------------------------------------------------
// Kernel 1: align_metric[b][a][g] for all pairs.
// tgt_sc gather = scores(16x80) x onehot(80x64) done with V_WMMA_F32_16X16X4_F32
// (exact: one-hot column has a single 1.0). Each wave: 16 anchors x 64 gts.
// Block = 256 thr = 8 waves = 128 anchors. Grid = (ceil(NA/128), BS).
// ---------------------------------------------------------------------------
__global__ __launch_bounds__(256) void k_am(
    const float* __restrict__ scores, const float* __restrict__ pbox,
    const float* __restrict__ anc,    const int*   __restrict__ labels,
    const float* __restrict__ gbox,   const float* __restrict__ gmask,
    float* __restrict__ am)
{
    __shared__ float sg[NB][8];    // gx1,gy1,gx2,gy2,area2,atan2,gm,_
    __shared__ int   sl[NB];
    __shared__ float sa[128][8];   // x1,y1,x2,y2,ax,ay,atan1,area1

    const int b  = blockIdx.y;
    const int a0 = blockIdx.x * 128;
    const int t  = threadIdx.x;

    if (t < NB) {
        const float* gb = gbox + ((size_t)b * NB + t) * 4;
        float gx1 = gb[0], gy1 = gb[1], gx2 = gb[2], gy2 = gb[3];
        float w2 = gx2 - gx1, h2 = gy2 - gy1 + 1e-7f;
        sg[t][0] = gx1; sg[t][1] = gy1; sg[t][2] = gx2; sg[t][3] = gy2;
        sg[t][4] = w2 * h2;
        sg[t][5] = atanf(w2 / h2);
        sg[t][6] = gmask[b * NB + t];
        sl[t]    = labels[b * NB + t];
    }
    if (t >= 128) {
        int i = t - 128;
        int a = a0 + i; if (a >= NA) a = NA - 1;
        const float* pb = pbox + ((size_t)b * NA + a) * 4;
        float x1 = pb[0], y1 = pb[1], x2 = pb[2], y2 = pb[3];
        float w1 = x2 - x1, h1 = y2 - y1 + 1e-7f;
        sa[i][0] = x1; sa[i][1] = y1; sa[i][2] = x2; sa[i][3] = y2;
        sa[i][4] = anc[a * 2 + 0];
        sa[i][5] = anc[a * 2 + 1];
        sa[i][6] = atanf(w1 / h1);
        sa[i][7] = w1 * h1;
    }
    __syncthreads();

    const int wave = t >> 5, lane = t & 31;
    const int lo = lane & 15, hi = lane >> 4;
    const int aw = a0 + wave * 16;        // wave's anchor-tile base

    // Labels for this lane's 4 N-tiles (gt = 16*j + lo).
    int lbl[4];
#pragma unroll
    for (int j = 0; j < 4; j++) lbl[j] = sl[16 * j + lo];

    // A-fragment source row: A[m][k], m = lo, k = k0 + 2*hi + {0,1}.
    int aload = aw + lo; if (aload >= NA) aload = NA - 1;
    const float* srow = scores + ((size_t)b * NA + aload) * NC + 2 * hi;

    v8f acc[4] = {};
#pragma unroll
    for (int k0 = 0; k0 < NC; k0 += 4) {
        v2f A = *(const v2f*)(srow + k0);
        const int kb = k0 + 2 * hi;
#pragma unroll
        for (int j = 0; j < 4; j++) {
            v2f Bf;
            Bf.x = (lbl[j] == kb)     ? 1.f : 0.f;
            Bf.y = (lbl[j] == kb + 1) ? 1.f : 0.f;
            acc[j] = __builtin_amdgcn_wmma_f32_16x16x4_f32(
                false, A, false, Bf, (short)0, acc[j], false, false);
        }
    }

    // Epilogue: lane holds tgt_sc for anchors m=v+8*hi, gts g=16*j+lo.
#pragma unroll
    for (int v = 0; v < 8; v++) {
        const int m = v + 8 * hi;
        const int a = aw + m;
        const float* A_ = sa[wave * 16 + m];
        float x1 = A_[0], y1 = A_[1], x2 = A_[2], y2 = A_[3];
        float ax = A_[4], ay = A_[5], at1 = A_[6], ar1 = A_[7];
#pragma unroll
        for (int j = 0; j < 4; j++) {
            const int g = 16 * j + lo;
            const float* G = sg[g];
            float gx1 = G[0], gy1 = G[1], gx2 = G[2], gy2 = G[3];
            float ar2 = G[4], at2 = G[5], gm = G[6];
            bool inbox = (ax - gx1 > EPSF) && (ay - gy1 > EPSF) &&
                         (gx2 - ax > EPSF) && (gy2 - ay > EPSF);
            float mp  = inbox ? gm : 0.f;
            float iou = ciou_clipped(x1, y1, x2, y2, ar1, at1,
                                     gx1, gy1, gx2, gy2, ar2, at2);
            float t2 = iou * iou;
            float i6 = t2 * t2 * t2;
            float tg = acc[j][v];
            float amv = sqrtf(fmaxf(tg, 0.f)) * i6 * mp;
            if (a < NA) am[((size_t)b * NA + a) * NB + g] = amv;
        }
    }
}

// ---------------------------------------------------------------------------
// Kernel 2: per (b,g) 10th-largest of am over anchors -> threshold T[b*64+g].
// Branchless register top-10 per thread + LDS tree merge. Grid = BS*NB blocks.
// ---------------------------------------------------------------------------
__global__ __launch_bounds__(256) void k_topk(const float* __restrict__ am,
                                              float* __restrict__ T)
{
    const int bg = blockIdx.x;
    const int b = bg >> 6, g = bg & 63;
    const int t = threadIdx.x;

    float top[KTOP];
#pragma unroll
    for (int i = 0; i < KTOP; i++) top[i] = -1.0f;   // am >= 0 always

    const float* col = am + ((size_t)b * NA) * NB + g;
    for (int a = t; a < NA; a += 256) {
        float v = col[(size_t)a * NB];
        __builtin_prefetch(&col[(size_t)(a + 256) * NB], 0, 0); // speculative
#pragma unroll
        for (int i = 0; i < KTOP; i++) {             // sorted-desc bubble insert
            float hiv = fmaxf(top[i], v);
            v = fminf(top[i], v);
            top[i] = hiv;
        }
    }

    __shared__ float s[256][KTOP];
#pragma unroll
    for (int i = 0; i < KTOP; i++) s[t][i] = top[i];
    __syncthreads();

    for (int stride = 128; stride >= 1; stride >>= 1) {
        if (t < stride) {
            float loc[KTOP];
#pragma unroll
            for (int i = 0; i < KTOP; i++) loc[i] = s[t][i];
#pragma unroll
            for (int q = 0; q < KTOP; q++) {
                float v = s[t + stride][q];
#pragma unroll
                for (int i = 0; i < KTOP; i++) {
                    float hiv = fmaxf(loc[i], v);
                    v = fminf(loc[i], v);
                    loc[i] = hiv;
                }
            }
#pragma unroll
            for (int i = 0; i < KTOP; i++) s[t][i] = loc[i];
        }
        __syncthreads();
    }
    if (t == 0) T[bg] = s[0][KTOP - 1];
}

// ---------------------------------------------------------------------------
// Kernel 3: per-anchor finalize. selected(g) = am>=T[g] && am>1e-9.
// conflict -> argmax gt; norm = amI^2/(iou_idx+1e-9) (single CIoU recompute).
// Output layout (floats): cls | bbox | scores | mask | norm.
// ---------------------------------------------------------------------------
__global__ __launch_bounds__(256) void k_final(
    const float* __restrict__ am,   const float* __restrict__ T,
    const int*   __restrict__ labels, const float* __restrict__ gbox,
    const float* __restrict__ pbox, float* __restrict__ out)
{
    __shared__ float sT[NB];
    __shared__ float sgb[NB][4];
    __shared__ int   sl[NB];

    const int b = blockIdx.y;
    const int t = threadIdx.x;
    if (t < NB) {
        sT[t] = T[b * NB + t];
        const float* gb = gbox + ((size_t)b * NB + t) * 4;
        sgb[t][0] = gb[0]; sgb[t][1] = gb[1]; sgb[t][2] = gb[2]; sgb[t][3] = gb[3];
        sl[t] = labels[b * NB + t];
    }
    __syncthreads();

    const int a = blockIdx.x * 256 + t;
    if (a >= NA) return;

    const float* row = am + ((size_t)b * NA + a) * NB;
    float bestV = -1.f, firstV = 0.f;
    int bestI = 0, firstI = -1, cnt = 0;
    for (int g = 0; g < NB; g++) {
        float v = row[g];
        if (v > bestV) { bestV = v; bestI = g; }
        if (v >= sT[g] && v > EPSF) {
            cnt++;
            if (firstI < 0) { firstI = g; firstV = v; }
        }
    }

    int idx; float amI; bool msk;
    if (cnt > 1)       { idx = bestI;  amI = bestV;  msk = true; }
    else if (cnt == 1) { idx = firstI; amI = firstV; msk = true; }
    else               { idx = 0;      amI = 0.f;    msk = false; }

    float gx1 = sgb[idx][0], gy1 = sgb[idx][1], gx2 = sgb[idx][2], gy2 = sgb[idx][3];
    int   cls = sl[idx];

    float normv = 0.f;
    if (msk) {
        const float* pb = pbox + ((size_t)b * NA + a) * 4;
        float x1 = pb[0], y1 = pb[1], x2 = pb[2], y2 = pb[3];
        float w1 = x2 - x1, h1 = y2 - y1 + 1e-7f;
        float w2 = gx2 - gx1, h2 = gy2 - gy1 + 1e-7f;
        float iou = ciou_clipped(x1, y1, x2, y2, w1 * h1, atanf(w1 / h1),
                                 gx1, gy1, gx2, gy2, w2 * h2, atanf(w2 / h2));
        normv = amI * amI / (iou + 1e-9f);
    }

    const size_t p = (size_t)b * NA + a;
    float* out_cls = out;
    float* out_box = out + (size_t)BS * NA;
    float* out_sc  = out + (size_t)BS * NA * 5;
    float* out_msk = out + (size_t)BS * NA * 85;
    float* out_nrm = out + (size_t)BS * NA * 86;

    out_cls[p] = (float)cls;
    out_box[p * 4 + 0] = gx1;
    out_box[p * 4 + 1] = gy1;
    out_box[p * 4 + 2] = gx2;
    out_box[p * 4 + 3] = gy2;

    float4* sc = (float4*)(out_sc + p * NC);
    float4  z  = make_float4(0.f, 0.f, 0.f, 0.f);
#pragma unroll
    for (int i = 0; i < NC / 4; i++) sc[i] = z;
    out_sc[p * NC + cls] = normv;

    out_msk[p] = msk ? 1.f : 0.f;
    out_nrm[p] = normv;
}

// ---------------------------------------------------------------------------
extern "C" void kernel_launch(void* const* d_in, const int* in_sizes, int n_in,
                              void* d_out, int out_size, void* d_ws, size_t ws_size,
                              hipStream_t stream)
{
    const float* scores = (const float*)d_in[0];   // [16,8400,80]
    const float* pbox   = (const float*)d_in[1];   // [16,8400,4]
    const float* anc    = (const float*)d_in[2];   // [8400,2]
    const int*   labels = (const int*)d_in[3];     // [16,64,1]
    const float* gbox   = (const float*)d_in[4];   // [16,64,4]
    const float* gmask  = (const float*)d_in[5];   // [16,64,1]

    float* am = (float*)d_ws;                      // 16*8400*64 floats (34.4 MB)
    float* T  = am + (size_t)BS * NA * NB;         // 1024 floats

    float* out = (float*)d_out;

    k_am  <<<dim3((NA + 127) / 128, BS), 256, 0, stream>>>(
        scores, pbox, anc, labels, gbox, gmask, am);
    k_topk<<<dim3(BS * NB), 256, 0, stream>>>(am, T);
    k_final<<<dim3((NA + 255) / 256, BS), 256, 0, stream>>>(
        am, T, labels, gbox, pbox, out);
}